// F1Loss_2336462209318
// MI455X (gfx1250) — compile-verified
//
#include <hip/hip_runtime.h>
#include <hip/hip_bf16.h>

// Problem constants (fixed by setup_inputs in the reference):
//   input  [N=8, C=19, H=512, W=1024] float32
//   target [N, H, W] integer (harness delivers as int32)
//   output: 1 scalar float = 1 - mean(f1 over N*C bins)

typedef float  f32x4 __attribute__((ext_vector_type(4)));
typedef int    i32x4 __attribute__((ext_vector_type(4)));
typedef float  v2f   __attribute__((ext_vector_type(2)));
typedef float  v8f   __attribute__((ext_vector_type(8)));

#define NBATCH 8
#define NCLASS 19
#define PLANE  (512 * 1024)       // H*W pixels per (n,c) plane
#define NBINS  (NCLASS * 3)       // tp | total_target | total_predict per image
#define NCNT   (NBATCH * NBINS)   // 456 u32 counters in d_ws
#define REP    4                  // LDS histogram replication factor

// ---------------------------------------------------------------------------
// Kernel 0: zero the global counters (d_ws is poisoned by the harness).
// ---------------------------------------------------------------------------
__global__ void f1_zero_kernel(unsigned int* __restrict__ cnt, int n) {
    int i = blockIdx.x * blockDim.x + threadIdx.x;
    if (i < n) cnt[i] = 0u;
}

// ---------------------------------------------------------------------------
// Kernel 1: streaming argmax + per-(n,class) counting.
//   grid = (PLANE / 1024, NBATCH), block = 256 threads, 4 pixels/thread.
//   Fully unrolled channel loop -> 19 x 128-bit non-temporal loads issued
//   back-to-back (~304 B in flight per thread) so HBM latency is hidden
//   within the wave, not just across waves. Replicated LDS histograms,
//   one global-atomic flush per block.
// ---------------------------------------------------------------------------
__global__ __launch_bounds__(256) void f1_count_kernel(
    const float* __restrict__ inp,
    const int*   __restrict__ tgt,
    unsigned int* __restrict__ cnt) {

    __shared__ unsigned int hist[REP][NBINS];

    const int tid = threadIdx.x;
    for (int j = tid; j < REP * NBINS; j += 256)
        (&hist[0][0])[j] = 0u;
    __syncthreads();

    const int    n     = blockIdx.y;
    const int    pix   = (blockIdx.x * 256 + tid) * 4;     // 16B-aligned
    const size_t nbase = (size_t)n * NCLASS * PLANE;

    // targets for 4 pixels
    const i32x4 t = *(const i32x4*)(tgt + (size_t)n * PLANE + pix);

    // Issue all 19 channel loads first (compiler clusters them), then the
    // compare chain consumes results in order.
    f32x4 v[NCLASS];
#pragma unroll
    for (int c = 0; c < NCLASS; ++c)
        v[c] = __builtin_nontemporal_load(
            (const f32x4*)(inp + nbase + (size_t)c * PLANE + pix));

    f32x4 bv = v[0];
    i32x4 bi = {0, 0, 0, 0};
#pragma unroll
    for (int c = 1; c < NCLASS; ++c) {
        // strict '>' keeps the first max -> matches jnp.argmax tie rule
        if (v[c].x > bv.x) { bv.x = v[c].x; bi.x = c; }
        if (v[c].y > bv.y) { bv.y = v[c].y; bi.y = c; }
        if (v[c].z > bv.z) { bv.z = v[c].z; bi.z = c; }
        if (v[c].w > bv.w) { bv.w = v[c].w; bi.w = c; }
    }

    unsigned int* h = hist[tid & (REP - 1)];
#pragma unroll
    for (int j = 0; j < 4; ++j) {
        const int pc = bi[j];
        const int tc = t[j];
        atomicAdd(&h[NCLASS + tc], 1u);        // total_target[tc]
        atomicAdd(&h[2 * NCLASS + pc], 1u);    // total_predict[pc]
        if (pc == tc) atomicAdd(&h[tc], 1u);   // tp[tc]
    }
    __syncthreads();

    // flush replicated histograms to global counters for image n
    for (int j = tid; j < NBINS; j += 256) {
        unsigned int s = hist[0][j] + hist[1][j] + hist[2][j] + hist[3][j];
        if (s) atomicAdd(&cnt[n * NBINS + j], s);
    }
}

// ---------------------------------------------------------------------------
// Kernel 2: epilogue. One wave (32 lanes, EXEC all ones).
//   Compute f1 for the 152 (n,c) bins in fp32, pad to 192, then reduce with
//   three chained V_WMMA_F32_16X16X4_F32 (B = ones => D accumulates exact
//   fp32 row sums; D column 0 lives in lanes 0 and 16 per the documented
//   16x16 f32 C/D layout).
// ---------------------------------------------------------------------------
__global__ void f1_final_kernel(const unsigned int* __restrict__ cnt,
                                float* __restrict__ out) {
    __shared__ float f1buf[192];
    __shared__ float red[32];

    const int   lane = threadIdx.x;          // 0..31
    const float S    = 1e-5f;

    for (int idx = lane; idx < 192; idx += 32) {   // uniform 6-iter loop
        float f1 = 0.0f;
        if (idx < NBATCH * NCLASS) {
            const int n = idx / NCLASS;
            const int c = idx % NCLASS;
            const float tp = (float)cnt[n * NBINS + c];
            const float tt = (float)cnt[n * NBINS + NCLASS + c];
            const float pp = (float)cnt[n * NBINS + 2 * NCLASS + c];
            const float r  = (tp + S) / (tt + S);
            const float p  = (tp + S) / (pp + S);
            f1 = 2.0f * r * p / (r + p);
        }
        f1buf[idx] = f1;
    }
    __syncthreads();

    // A layout (16x4 f32): lanes 0-15 -> K=0,1 ; lanes 16-31 -> K=2,3
    const int m  = lane & 15;
    const int kb = (lane >> 4) * 2;

    v8f acc = {};
    v2f b;  b.x = 1.0f; b.y = 1.0f;          // B = all ones (layout-free)

#pragma unroll
    for (int ch = 0; ch < 3; ++ch) {
        const float* base = f1buf + ch * 64 + m * 4 + kb;  // A[m][k]
        v2f a;  a.x = base[0];  a.y = base[1];
        // D = A x ones + C  ->  per-row exact fp32 accumulation
        acc = __builtin_amdgcn_wmma_f32_16x16x4_f32(
            false, a, false, b, (short)0, acc, false, false);
    }

    // D[m][0]: lanes 0 (M=0..7 in acc[0..7]) and 16 (M=8..15)
    float partial = acc[0] + acc[1] + acc[2] + acc[3] +
                    acc[4] + acc[5] + acc[6] + acc[7];
    red[lane] = partial;
    __syncthreads();

    if (lane == 0) {
        const float total = red[0] + red[16];             // sum of 152 f1's
        out[0] = 1.0f - total / (float)(NBATCH * NCLASS); // 1 - mean(f1)
    }
}

// ---------------------------------------------------------------------------
extern "C" void kernel_launch(void* const* d_in, const int* in_sizes, int n_in,
                              void* d_out, int out_size, void* d_ws, size_t ws_size,
                              hipStream_t stream) {
    const float* inp = (const float*)d_in[0];
    const int*   tgt = (const int*)d_in[1];
    float*       out = (float*)d_out;
    unsigned int* cnt = (unsigned int*)d_ws;   // 456 u32 counters

    f1_zero_kernel<<<1, 512, 0, stream>>>(cnt, NCNT);

    dim3 grid(PLANE / (256 * 4), NBATCH);      // (512, 8)
    f1_count_kernel<<<grid, 256, 0, stream>>>(inp, tgt, cnt);

    f1_final_kernel<<<1, 32, 0, stream>>>(cnt, out);
}